// MinLSTMModel_59923383714529
// MI455X (gfx1250) — compile-verified
//
#include <hip/hip_runtime.h>

typedef __attribute__((ext_vector_type(2))) float v2f;
typedef __attribute__((ext_vector_type(8))) float v8f;

// ---------------------------------------------------------------------------
// gfx1250 async global->LDS copy (ASYNCcnt path). Builtin prototype (from the
// compiler diagnostic): param0 = int4 ptr in AS(1) (non-const), 4 args total.
// ---------------------------------------------------------------------------
#if __has_builtin(__builtin_amdgcn_global_load_async_to_lds_b128)
#define USE_ASYNC 1
#else
#define USE_ASYNC 0
#endif

#if USE_ASYNC
#if __has_builtin(__builtin_amdgcn_s_wait_asynccnt)
#define WAIT_ASYNC0() __builtin_amdgcn_s_wait_asynccnt(0)
#else
#define WAIT_ASYNC0() asm volatile("s_wait_asynccnt 0" ::: "memory")
#endif
typedef int v4i_t __attribute__((__vector_size__(4 * sizeof(int))));
typedef __attribute__((address_space(1))) v4i_t AS1_v4i;
typedef __attribute__((address_space(3))) v4i_t AS3_v4i;
__device__ __forceinline__ void async_cp16(const void* g, void* l) {
  // 16B per lane, memory -> LDS, no VGPR data path. AS3 pointer value is the
  // wave-relative LDS byte offset = low 32 bits of the generic address.
  __builtin_amdgcn_global_load_async_to_lds_b128(
      (AS1_v4i*)(uintptr_t)g, (AS3_v4i*)(unsigned)(uintptr_t)l, 0, 0);
}
#endif

namespace {

constexpr int kT = 1024, kB = 64, kIn = 256, kLstm = 1024, kFnn = 4096, kOut = 256;
constexpr int kM = kT * kB; // 65536 rows for every GEMM

__device__ __forceinline__ float fast_sigmoid(float x) {
  return 1.0f / (1.0f + __expf(-x));
}
__device__ __forceinline__ float gelu_exact(float x) {
  return 0.5f * x * (1.0f + erff(x * 0.70710678118654752440f));
}

// ---------------------------------------------------------------------------
// Kernel 1: fused 3-gate projection GEMM (x read once for Wf/Wi/Wh).
// Block = 256 threads (8 waves), tile 128(M) x 64(N), K chunk 16,
// double-buffered LDS, async A staging, pair-interleaved B tiles.
// Waves 4(M) x 2(N); each wave owns 2x2 16x16 tiles per weight (96 acc VGPRs).
// ---------------------------------------------------------------------------
__global__ __launch_bounds__(256)
void gates_kernel(const float* __restrict__ X,
                  const float* __restrict__ Wf, const float* __restrict__ Wi,
                  const float* __restrict__ Wh,
                  const float* __restrict__ bf, const float* __restrict__ bi,
                  const float* __restrict__ bh,
                  float* __restrict__ aOut, float* __restrict__ bOut)
{
  constexpr int BM = 128, BN = 64, KC = 16, KP = KC / 2, K = kIn, N = kLstm;
  constexpr int SAW = KC + 4;  // 20: rows 16B-aligned for async, frag-conflict-free
  constexpr int SBW = BN + 16; // pair-row stride 160 floats == 32 mod 64 banks
  __shared__ float sA[2][BM][SAW];
  __shared__ float sB[2][3][KP][SBW][2]; // [k/2][n][k&1] -> B frag = one b64

  const int tid  = threadIdx.x;
  const int lane = tid & 31;
  const int wv   = tid >> 5;
  const int lm   = lane & 15;
  const int lhi  = lane >> 4;
  const int m0   = blockIdx.x * BM;
  const int n0   = blockIdx.y * BN;
  const int wm   = (wv & 3) * 32;
  const int wn   = (wv >> 2) * 32;

  // staging coordinates: A = 2 x b128/thread, B = 1 x b128/thread/weight
  const int arA[2] = {tid >> 2, (256 + tid) >> 2};
  const int acA    = (tid & 3) * 4;
  const int rB     = tid >> 4;
  const int cB     = (tid & 15) * 4;
  const float* const Ws[3] = {Wf, Wi, Wh};

  v8f accF[2][2] = {}; v8f accI[2][2] = {}; v8f accH[2][2] = {};
  float4 aReg[2], bReg[3];

  // ---- prologue: stage chunk 0 ----
#pragma unroll
  for (int rep = 0; rep < 2; ++rep) {
#if USE_ASYNC
    async_cp16(X + (size_t)(m0 + arA[rep]) * K + acA, &sA[0][arA[rep]][acA]);
#else
    aReg[rep] = *(const float4*)(X + (size_t)(m0 + arA[rep]) * K + acA);
#endif
  }
#pragma unroll
  for (int w = 0; w < 3; ++w)
    bReg[w] = *(const float4*)(Ws[w] + (size_t)rB * N + n0 + cB);
#pragma unroll
  for (int w = 0; w < 3; ++w) {
    sB[0][w][rB >> 1][cB + 0][rB & 1] = bReg[w].x;
    sB[0][w][rB >> 1][cB + 1][rB & 1] = bReg[w].y;
    sB[0][w][rB >> 1][cB + 2][rB & 1] = bReg[w].z;
    sB[0][w][rB >> 1][cB + 3][rB & 1] = bReg[w].w;
  }
#if USE_ASYNC
  WAIT_ASYNC0();
#else
#pragma unroll
  for (int rep = 0; rep < 2; ++rep) {
    sA[0][arA[rep]][acA + 0] = aReg[rep].x; sA[0][arA[rep]][acA + 1] = aReg[rep].y;
    sA[0][arA[rep]][acA + 2] = aReg[rep].z; sA[0][arA[rep]][acA + 3] = aReg[rep].w;
  }
#endif
  __syncthreads();

  for (int kc = 0; kc < K; kc += KC) {
    const int cur  = (kc >> 4) & 1;
    const int nxt  = cur ^ 1;
    const bool more = (kc + KC) < K;

    if (more) { // issue next chunk's traffic before computing
#pragma unroll
      for (int w = 0; w < 3; ++w)
        bReg[w] = *(const float4*)(Ws[w] + (size_t)(kc + KC + rB) * N + n0 + cB);
#pragma unroll
      for (int rep = 0; rep < 2; ++rep) {
#if USE_ASYNC
        async_cp16(X + (size_t)(m0 + arA[rep]) * K + kc + KC + acA,
                   &sA[nxt][arA[rep]][acA]);
#else
        aReg[rep] = *(const float4*)(X + (size_t)(m0 + arA[rep]) * K + kc + KC + acA);
#endif
      }
    }

    // ---- WMMA on current buffers ----
#pragma unroll
    for (int k0 = 0; k0 < KC; k0 += 4) {
      const int kb = k0 + 2 * lhi;
      v2f af[2];
#pragma unroll
      for (int tm = 0; tm < 2; ++tm)
        af[tm] = *(const v2f*)&sA[cur][wm + tm * 16 + lm][kb];
#pragma unroll
      for (int tn = 0; tn < 2; ++tn) {
        const int nl = wn + tn * 16 + lm;
        const v2f bF = *(const v2f*)&sB[cur][0][kb >> 1][nl][0];
        const v2f bI = *(const v2f*)&sB[cur][1][kb >> 1][nl][0];
        const v2f bH = *(const v2f*)&sB[cur][2][kb >> 1][nl][0];
#pragma unroll
        for (int tm = 0; tm < 2; ++tm) {
          accF[tm][tn] = __builtin_amdgcn_wmma_f32_16x16x4_f32(
              false, af[tm], false, bF, (short)0, accF[tm][tn], false, false);
          accI[tm][tn] = __builtin_amdgcn_wmma_f32_16x16x4_f32(
              false, af[tm], false, bI, (short)0, accI[tm][tn], false, false);
          accH[tm][tn] = __builtin_amdgcn_wmma_f32_16x16x4_f32(
              false, af[tm], false, bH, (short)0, accH[tm][tn], false, false);
        }
      }
    }

    if (more) { // drain regs into the other buffer while WMMA retires
#pragma unroll
      for (int w = 0; w < 3; ++w) {
        sB[nxt][w][rB >> 1][cB + 0][rB & 1] = bReg[w].x;
        sB[nxt][w][rB >> 1][cB + 1][rB & 1] = bReg[w].y;
        sB[nxt][w][rB >> 1][cB + 2][rB & 1] = bReg[w].z;
        sB[nxt][w][rB >> 1][cB + 3][rB & 1] = bReg[w].w;
      }
#if USE_ASYNC
      WAIT_ASYNC0();
#else
#pragma unroll
      for (int rep = 0; rep < 2; ++rep) {
        sA[nxt][arA[rep]][acA + 0] = aReg[rep].x; sA[nxt][arA[rep]][acA + 1] = aReg[rep].y;
        sA[nxt][arA[rep]][acA + 2] = aReg[rep].z; sA[nxt][arA[rep]][acA + 3] = aReg[rep].w;
      }
#endif
    }
    __syncthreads();
  }

  // epilogue: bias + sigmoid + gate combine -> a, b coefficient streams
#pragma unroll
  for (int tn = 0; tn < 2; ++tn) {
    const int n = n0 + wn + tn * 16 + lm;
    const float bfv = bf[n], biv = bi[n], bhv = bh[n];
#pragma unroll
    for (int tm = 0; tm < 2; ++tm) {
#pragma unroll
      for (int v = 0; v < 8; ++v) {
        const int m = m0 + wm + tm * 16 + v + 8 * lhi;
        const float fg = fast_sigmoid(accF[tm][tn][v] + bfv);
        const float ig = fast_sigmoid(accI[tm][tn][v] + biv);
        const float hh = accH[tm][tn][v] + bhv;
        const float s  = fg + ig;
        const size_t o = (size_t)m * N + n;
        aOut[o] = fg / s;
        bOut[o] = (ig / s) * hh;
      }
    }
  }
}

// ---------------------------------------------------------------------------
// Kernel 2: linear recurrence h_t = a_t*h_{t-1} + b_t, T=1024 steps.
// One thread per (batch, lstm) lane, coalesced; h written over `a` in place.
// ---------------------------------------------------------------------------
__global__ __launch_bounds__(256)
void scan_kernel(float* __restrict__ AB, const float* __restrict__ Bc)
{
  const int idx = blockIdx.x * blockDim.x + threadIdx.x; // 0 .. B*LSTM-1
  float h = 0.0f;
  for (int t = 0; t < kT; ++t) {
    const size_t o = (size_t)t * (kB * kLstm) + idx;
    h = AB[o] * h + Bc[o];
    AB[o] = h;
  }
}

// ---------------------------------------------------------------------------
// Kernels 3/4: generic fp32 WMMA GEMM, tile 128x128, K chunk 16,
// double-buffered + async A staging + pair-interleaved B.
// 8 waves 4(M) x 2(N); each wave owns 2x4 16x16 tiles (64 acc VGPRs).
// EPI==0: GELU epilogue; EPI==1: bias only.
// ---------------------------------------------------------------------------
template <int EPI>
__global__ __launch_bounds__(256)
void wmma_gemm_f32(const float* __restrict__ A, const float* __restrict__ W,
                   const float* __restrict__ bias, float* __restrict__ Out,
                   const int K, const int N)
{
  constexpr int BM = 128, BN = 128, KC = 16, KP = KC / 2;
  constexpr int SAW = KC + 4;  // 20
  constexpr int SBW = BN + 16; // 144: pair-row stride 288 floats == 32 mod 64
  __shared__ float sA[2][BM][SAW];
  __shared__ float sB[2][KP][SBW][2];

  const int tid  = threadIdx.x;
  const int lane = tid & 31;
  const int wv   = tid >> 5;
  const int lm   = lane & 15;
  const int lhi  = lane >> 4;
  const int m0   = blockIdx.x * BM;
  const int n0   = blockIdx.y * BN;
  const int wm   = (wv & 3) * 32;
  const int wn   = (wv >> 2) * 64;

  const int arA[2] = {tid >> 2, (256 + tid) >> 2};
  const int acA    = (tid & 3) * 4;
  const int brB[2] = {tid >> 5, (256 + tid) >> 5};
  const int bcB    = (tid & 31) * 4;

  v8f acc[2][4] = {};
  float4 aReg[2], bReg[2];

  // ---- prologue: stage chunk 0 ----
#pragma unroll
  for (int rep = 0; rep < 2; ++rep) {
#if USE_ASYNC
    async_cp16(A + (size_t)(m0 + arA[rep]) * K + acA, &sA[0][arA[rep]][acA]);
#else
    aReg[rep] = *(const float4*)(A + (size_t)(m0 + arA[rep]) * K + acA);
#endif
    bReg[rep] = *(const float4*)(W + (size_t)brB[rep] * N + n0 + bcB);
  }
#pragma unroll
  for (int rep = 0; rep < 2; ++rep) {
    const int r = brB[rep];
    sB[0][r >> 1][bcB + 0][r & 1] = bReg[rep].x;
    sB[0][r >> 1][bcB + 1][r & 1] = bReg[rep].y;
    sB[0][r >> 1][bcB + 2][r & 1] = bReg[rep].z;
    sB[0][r >> 1][bcB + 3][r & 1] = bReg[rep].w;
#if !USE_ASYNC
    sA[0][arA[rep]][acA + 0] = aReg[rep].x; sA[0][arA[rep]][acA + 1] = aReg[rep].y;
    sA[0][arA[rep]][acA + 2] = aReg[rep].z; sA[0][arA[rep]][acA + 3] = aReg[rep].w;
#endif
  }
#if USE_ASYNC
  WAIT_ASYNC0();
#endif
  __syncthreads();

  for (int kc = 0; kc < K; kc += KC) {
    const int cur  = (kc >> 4) & 1;
    const int nxt  = cur ^ 1;
    const bool more = (kc + KC) < K;

    if (more) {
#pragma unroll
      for (int rep = 0; rep < 2; ++rep) {
        bReg[rep] = *(const float4*)(W + (size_t)(kc + KC + brB[rep]) * N + n0 + bcB);
#if USE_ASYNC
        async_cp16(A + (size_t)(m0 + arA[rep]) * K + kc + KC + acA,
                   &sA[nxt][arA[rep]][acA]);
#else
        aReg[rep] = *(const float4*)(A + (size_t)(m0 + arA[rep]) * K + kc + KC + acA);
#endif
      }
    }

#pragma unroll
    for (int k0 = 0; k0 < KC; k0 += 4) {
      const int kb = k0 + 2 * lhi;
      v2f af[2];
#pragma unroll
      for (int tm = 0; tm < 2; ++tm)
        af[tm] = *(const v2f*)&sA[cur][wm + tm * 16 + lm][kb];
#pragma unroll
      for (int tn = 0; tn < 4; ++tn) {
        const v2f bfr = *(const v2f*)&sB[cur][kb >> 1][wn + tn * 16 + lm][0];
#pragma unroll
        for (int tm = 0; tm < 2; ++tm)
          acc[tm][tn] = __builtin_amdgcn_wmma_f32_16x16x4_f32(
              false, af[tm], false, bfr, (short)0, acc[tm][tn], false, false);
      }
    }

    if (more) {
#pragma unroll
      for (int rep = 0; rep < 2; ++rep) {
        const int r = brB[rep];
        sB[nxt][r >> 1][bcB + 0][r & 1] = bReg[rep].x;
        sB[nxt][r >> 1][bcB + 1][r & 1] = bReg[rep].y;
        sB[nxt][r >> 1][bcB + 2][r & 1] = bReg[rep].z;
        sB[nxt][r >> 1][bcB + 3][r & 1] = bReg[rep].w;
#if !USE_ASYNC
        sA[nxt][arA[rep]][acA + 0] = aReg[rep].x; sA[nxt][arA[rep]][acA + 1] = aReg[rep].y;
        sA[nxt][arA[rep]][acA + 2] = aReg[rep].z; sA[nxt][arA[rep]][acA + 3] = aReg[rep].w;
#endif
      }
#if USE_ASYNC
      WAIT_ASYNC0();
#endif
    }
    __syncthreads();
  }

#pragma unroll
  for (int tn = 0; tn < 4; ++tn) {
    const int n = n0 + wn + tn * 16 + lm;
    const float bv = bias[n];
#pragma unroll
    for (int tm = 0; tm < 2; ++tm) {
#pragma unroll
      for (int v = 0; v < 8; ++v) {
        const int m = m0 + wm + tm * 16 + v + 8 * lhi;
        float x = acc[tm][tn][v] + bv;
        if (EPI == 0) x = gelu_exact(x);
        Out[(size_t)m * N + n] = x;
      }
    }
  }
}

} // namespace

extern "C" void kernel_launch(void* const* d_in, const int* in_sizes, int n_in,
                              void* d_out, int out_size, void* d_ws, size_t ws_size,
                              hipStream_t stream) {
  (void)in_sizes; (void)n_in; (void)out_size;
  const float* x  = (const float*)d_in[0];
  const float* Wf = (const float*)d_in[1];
  const float* bf = (const float*)d_in[2];
  const float* Wi = (const float*)d_in[3];
  const float* bi = (const float*)d_in[4];
  const float* Wh = (const float*)d_in[5];
  const float* bh = (const float*)d_in[6];
  const float* W1 = (const float*)d_in[7];
  const float* b1 = (const float*)d_in[8];
  const float* W2 = (const float*)d_in[9];
  const float* b2 = (const float*)d_in[10];
  float* out = (float*)d_out;

  const size_t nMH = (size_t)kM * kLstm;  // a/b coef planes (T*B, LSTM)
  const size_t nMF = (size_t)kM * kFnn;   // gelu activation plane
  if (ws_size < (2 * nMH + nMF) * sizeof(float)) return;  // insufficient scratch
  float* aBuf = (float*)d_ws;      // a coefs, later overwritten by h
  float* bBuf = aBuf + nMH;        // b coefs
  float* gBuf = bBuf + nMH;        // gelu(h@W1+b1)

  gates_kernel<<<dim3(kM / 128, kLstm / 64), 256, 0, stream>>>(
      x, Wf, Wi, Wh, bf, bi, bh, aBuf, bBuf);
  scan_kernel<<<dim3((kB * kLstm) / 256), 256, 0, stream>>>(aBuf, bBuf);
  wmma_gemm_f32<0><<<dim3(kM / 128, kFnn / 128), 256, 0, stream>>>(
      aBuf, W1, b1, gBuf, kLstm, kFnn);
  wmma_gemm_f32<1><<<dim3(kM / 128, kOut / 128), 256, 0, stream>>>(
      gBuf, W2, b2, out, kFnn, kOut);
}